// FullBCModel_7481833030196
// MI455X (gfx1250) — compile-verified
//
#include <hip/hip_runtime.h>
#include <hip/hip_bf16.h>
#include <math.h>

// ---------------- model constants ----------------
#define T_STIM   32768
#define NUM_PR   2
#define NUM_BC   14
#define NUM_ACL  45
#define K0       20                      // BC kernel taps
#define KAC      52                      // AC kernel taps
#define K1       64                      // sensor kernel taps
#define PADLEN   (K0 - 1 + K1 - 1)       // 82
#define TP       (T_STIM + K1 - 1)       // 32831 scan steps
#define DRV_STRIDE 16
#define FREQF    64.0f
#define TWO_PI_F 6.283185307179586f

// ---------------- workspace layout (float offsets) ----------------
#define OFF_KMAT   0            // 40 x 16 (K x N, N padded 14->16, zeros)
#define OFF_ACK    640          // 45 x 52
#define OFF_WBA    2980         // 45 x 14
#define OFF_WAA    3610         // 45 x 45 (already negated)
#define OFF_WAB    5635         // 14 x 45 (already negated)
#define OFF_SLB    6265         // 14
#define OFF_OFB    6279
#define OFF_P01    6293
#define OFF_P12    6307
#define OFF_IPC    6321
#define OFF_RRC    6335
#define OFF_IP0    6349
#define OFF_RRP0   6363
#define OFF_SLA    6377         // 45
#define OFF_OFA    6422         // 45
#define OFF_IGLU   6467         // 64
#define OFF_GCAMP  6531         // 64
#define OFF_WCB    6595         // 14 x 45: W_ab + W_ab @ W_aa (critical-path fold)
#define OFF_DRIVE  7232         // TP x 16
#define OFF_ACL    532528       // TP x 45   (ends ~2,009,923 floats ~ 8 MB)

// d_out layout: y_lnr (T x 14) | rel (TP x 14) | y_calcium (T x 45)
#define OUT_REL_OFF  ((size_t)T_STIM * NUM_BC)
#define OUT_YC_OFF   ((size_t)T_STIM * NUM_BC + (size_t)TP * NUM_BC)

typedef float v2f __attribute__((ext_vector_type(2)));
typedef float v8f __attribute__((ext_vector_type(8)));

// gfx1250 async global->LDS path (ASYNCcnt), guarded so a missing builtin
// falls back to plain loads instead of failing the build.
// Probe round 2 showed the builtin's params are (AS1 int*, AS3 int*, imm, imm).
#if defined(__has_builtin)
#if __has_builtin(__builtin_amdgcn_global_load_async_to_lds_b32)
#define HAVE_ASYNC_LDS 1
#define ASYNC_CP_F32(gsrc, ldst)                                              \
    __builtin_amdgcn_global_load_async_to_lds_b32(                            \
        (__attribute__((address_space(1))) int*)(gsrc),                       \
        (__attribute__((address_space(3))) int*)(ldst), 0, 0)
#if __has_builtin(__builtin_amdgcn_s_wait_asynccnt)
#define WAIT_ASYNC() __builtin_amdgcn_s_wait_asynccnt(0)
#else
#define WAIT_ASYNC() asm volatile("s_wait_asynccnt 0x0" ::: "memory")
#endif
#endif
#endif

// =================================================================
// Kernel 0: precompute all static tables from the parameters.
// =================================================================
__global__ __launch_bounds__(128) void precompute_kernel(
    const float* __restrict__ lw,    // log_pr_bc_weight (14,2)
    const float* __restrict__ lks,   // log_kernel_speed (14,2)
    const float* __restrict__ soff,  // sigmoid_offset (14)
    const float* __restrict__ lss,   // log_sigmoid_slope (14)
    const float* __restrict__ lcp01, const float* __restrict__ lcp12,
    const float* __restrict__ lic,   const float* __restrict__ lrc,
    const float* __restrict__ sips,  const float* __restrict__ srps,
    const float* __restrict__ lba,   // log_bc_acl_weight (45,14)
    const float* __restrict__ ltr,   const float* __restrict__ ltd,
    const float* __restrict__ lass,  const float* __restrict__ aso,
    const float* __restrict__ lab,   // log_acl_bc_weight (14,45)
    const float* __restrict__ laa,   // log_acl_acl_weight (45,45)
    float* __restrict__ ws)
{
    const int tid = threadIdx.x;

    for (int i = tid; i < 40 * 16; i += blockDim.x) ws[OFF_KMAT + i] = 0.0f;
    __syncthreads();

    // biphasic BC kernels, L2-normalized, folded with exp(log_pr_bc_weight)
    if (tid < NUM_BC * NUM_PR) {
        const int bc = tid >> 1, pr = tid & 1;
        const float s = __expf(lks[bc * NUM_PR + pr]);
        float v[K0]; float nrm = 0.0f;
        for (int k = 0; k < K0; ++k) {
            const float t = (k / FREQF) * s;
            const float val = __expf(-t / 0.04f) * __cosf(TWO_PI_F * t / 0.24f);
            v[k] = val; nrm += val * val;
        }
        nrm = 1.0f / sqrtf(nrm);
        const float w = __expf(lw[bc * NUM_PR + pr]);
        for (int k = 0; k < K0; ++k)
            ws[OFF_KMAT + (pr * K0 + k) * 16 + bc] = w * v[k] * nrm;
    }

    // AC double-exponential kernels (oldest-first time axis), L2-normalized
    if (tid < NUM_ACL) {
        const float tr = __expf(ltr[tid]), td = __expf(ltd[tid]);
        const float c = (td + tr) / (td * tr);
        float v[KAC]; float nrm = 0.0f;
        for (int j = 0; j < KAC; ++j) {
            const float kt = 0.8f - j / FREQF;
            const float val = __expf(-kt / td) - __expf(-c * kt);
            v[j] = val; nrm += val * val;
        }
        nrm = 1.0f / sqrtf(nrm);
        for (int j = 0; j < KAC; ++j) ws[OFF_ACK + tid * KAC + j] = v[j] * nrm;
        ws[OFF_SLA + tid] = __expf(lass[tid]);
        ws[OFF_OFA + tid] = aso[tid];
    }

    if (tid < NUM_BC) {
        ws[OFF_SLB + tid] = __expf(lss[tid]);
        ws[OFF_OFB + tid] = soff[tid];
        const float p01 = __expf(lcp01[tid]), p12 = __expf(lcp12[tid]);
        const float ipc = __expf(lic[tid]),   rrc = __expf(lrc[tid]);
        ws[OFF_P01 + tid] = p01;  ws[OFF_P12 + tid] = p12;
        ws[OFF_IPC + tid] = ipc;  ws[OFF_RRC + tid] = rrc;
        ws[OFF_IP0  + tid] = ipc / (1.0f + __expf(-sips[tid]));
        ws[OFF_RRP0 + tid] = rrc / (1.0f + __expf(-srps[tid]));
    }

    for (int i = tid; i < NUM_ACL * NUM_BC;  i += blockDim.x) ws[OFF_WBA + i] =  __expf(lba[i]);
    for (int i = tid; i < NUM_ACL * NUM_ACL; i += blockDim.x) ws[OFF_WAA + i] = -__expf(laa[i]);
    for (int i = tid; i < NUM_BC * NUM_ACL;  i += blockDim.x) ws[OFF_WAB + i] = -__expf(lab[i]);

    if (tid == 0) {     // iGluSnFR kernel: _sensor_kernel(0.01, 0.1)
        float v[K1]; float nrm = 0.0f;
        for (int k = 0; k < K1; ++k) {
            const float t = k / FREQF;
            const float val = __expf(-t / 0.1f) - __expf(-t / 0.01f);
            v[k] = val; nrm += val * val;
        }
        nrm = 1.0f / sqrtf(nrm);
        for (int k = 0; k < K1; ++k) ws[OFF_IGLU + k] = v[k] * nrm;
    }
    if (tid == 1) {     // GCaMP6f kernel: _sensor_kernel(0.05, 0.4)
        float v[K1]; float nrm = 0.0f;
        for (int k = 0; k < K1; ++k) {
            const float t = k / FREQF;
            const float val = __expf(-t / 0.4f) - __expf(-t / 0.05f);
            v[k] = val; nrm += val * val;
        }
        nrm = 1.0f / sqrtf(nrm);
        for (int k = 0; k < K1; ++k) ws[OFF_GCAMP + k] = v[k] * nrm;
    }

    // Wcomb = W_ab + W_ab @ W_aa  (14x45): removes the 45x45 matvec from
    // the scan's critical path (feedback = Wcomb @ acl_sigmoid, exactly).
    __syncthreads();
    for (int i = tid; i < NUM_BC * NUM_ACL; i += blockDim.x) {
        const int b = i / NUM_ACL, a = i % NUM_ACL;
        float s = ws[OFF_WAB + b * NUM_ACL + a];
        for (int a2 = 0; a2 < NUM_ACL; ++a2)
            s = fmaf(ws[OFF_WAB + b * NUM_ACL + a2], ws[OFF_WAA + a2 * NUM_ACL + a], s);
        ws[OFF_WCB + i] = s;
    }
}

// =================================================================
// Kernel 1: drive = Window(x) (TPx40) @ Kmat (40x16) via f32 WMMA.
// One 16x16 output tile per wave; K=40 -> 10 x V_WMMA_F32_16X16X4_F32.
// =================================================================
__global__ __launch_bounds__(256) void drive_wmma_kernel(
    const float* __restrict__ x, const float* __restrict__ sbias,
    const float* __restrict__ sscale, float* __restrict__ ws)
{
    const float scale = sscale[0], bias = sbias[0];
    const int lane = threadIdx.x & 31;
    const int wave = threadIdx.x >> 5;
    const int tile = blockIdx.x * 8 + wave;
    const int ntiles = (TP + 15) / 16;
    if (tile >= ntiles) return;                 // wave-uniform: EXEC stays all-1s

    const int i0   = tile * 16;
    const int m    = lane & 15;                 // A row / B col within tile
    const int koff = (lane >> 4) << 1;          // 0 or 2
    const float* __restrict__ kmat = ws + OFF_KMAT;
    float* __restrict__ drive = ws + OFF_DRIVE;

    v8f acc = {};
    #pragma unroll
    for (int t = 0; t < 10; ++t) {
        const int kb = t * 4 + koff;            // this lane's first K column
        v2f a, b;
        // A window element: A[i, c] with c = pr*20 + k  ->  xs_pr[i + k],
        // xs_pr[j] = x[clamp(j-82), pr]*scale + bias (left pad with x[0]).
        {
            const int c0 = kb, c1 = kb + 1;
            const int pr0 = c0 / K0, k0 = c0 - pr0 * K0;
            const int pr1 = c1 / K0, k1 = c1 - pr1 * K0;
            int j0 = i0 + m + k0 - PADLEN; j0 = min(max(j0, 0), T_STIM - 1);
            int j1 = i0 + m + k1 - PADLEN; j1 = min(max(j1, 0), T_STIM - 1);
            a.x = x[j0 * NUM_PR + pr0] * scale + bias;
            a.y = x[j1 * NUM_PR + pr1] * scale + bias;
        }
        b.x = kmat[(kb    ) * 16 + m];
        b.y = kmat[(kb + 1) * 16 + m];
        acc = __builtin_amdgcn_wmma_f32_16x16x4_f32(
            false, a, false, b, (short)0, acc, false, false);
    }
    #pragma unroll
    for (int r = 0; r < 8; ++r) {
        const int row = i0 + r + ((lane >> 4) << 3);
        if (row < TP) drive[(size_t)row * DRV_STRIDE + m] = acc[r];
    }
}

// =================================================================
// Kernel 2: the inherently serial scan. One block, 64 threads,
// 3 barriers/step. History kept as g[a,t] = W_ba @ rel[t] in a
// 52-deep LDS ring. Feedback uses the precomputed Wcomb so the
// aclO emission (threads 14..58) runs concurrently with the BC
// release update (threads 0..13). 4-way accumulators break the
// dependent-FMA chains.
// =================================================================
__global__ __launch_bounds__(64) void scan_kernel(
    float* __restrict__ ws, float* __restrict__ out)
{
    __shared__ float ackS[NUM_ACL][KAC];
    __shared__ float wbaS[NUM_ACL][NUM_BC];
    __shared__ float waaS[NUM_ACL][NUM_ACL];
    __shared__ float wcbS[NUM_BC][NUM_ACL];
    __shared__ float gring[KAC][NUM_ACL + 3];   // ring of g values, padded stride
    __shared__ float aclS[NUM_ACL];
    __shared__ float relS[NUM_BC];

    const int tid = threadIdx.x;
    for (int i = tid; i < NUM_ACL * KAC;     i += 64) ackS[i / KAC][i % KAC]         = ws[OFF_ACK + i];
    for (int i = tid; i < NUM_ACL * NUM_BC;  i += 64) wbaS[i / NUM_BC][i % NUM_BC]   = ws[OFF_WBA + i];
    for (int i = tid; i < NUM_ACL * NUM_ACL; i += 64) waaS[i / NUM_ACL][i % NUM_ACL] = ws[OFF_WAA + i];
    for (int i = tid; i < NUM_BC * NUM_ACL;  i += 64) wcbS[i / NUM_ACL][i % NUM_ACL] = ws[OFF_WCB + i];
    for (int i = tid; i < KAC * (NUM_ACL + 3); i += 64) (&gring[0][0])[i] = 0.0f;
    __syncthreads();

    float slA = 0.f, ofA = 0.f;
    if (tid < NUM_ACL) { slA = ws[OFF_SLA + tid]; ofA = ws[OFF_OFA + tid]; }
    float slB = 0.f, ofB = 0.f, p01 = 0.f, p12 = 0.f, ipc = 1.f, rrc = 1.f, ip = 0.f, rrp = 0.f;
    if (tid < NUM_BC) {
        slB = ws[OFF_SLB + tid]; ofB = ws[OFF_OFB + tid];
        p01 = ws[OFF_P01 + tid]; p12 = ws[OFF_P12 + tid];
        ipc = ws[OFF_IPC + tid]; rrc = ws[OFF_RRC + tid];
        ip  = ws[OFF_IP0 + tid]; rrp = ws[OFF_RRP0 + tid];
    }
    const int aout = tid - NUM_BC;              // ACL id for the aclO side-job
    const float* __restrict__ drive = ws + OFF_DRIVE;
    float* __restrict__ aclout = ws + OFF_ACL;
    float* __restrict__ relout = out + OUT_REL_OFF;

    for (int i = 0; i < TP; ++i) {
        const int base = i % KAC;               // slot of oldest entry (time i-52)
        // P1: AC drive = 52-tap conv of g history; sigmoid
        if (tid < NUM_ACL) {
            float s0 = 0.f, s1 = 0.f, s2 = 0.f, s3 = 0.f;
            #pragma unroll
            for (int j = 0; j < KAC; j += 4) {
                int q0 = base + j;     q0 -= (q0 >= KAC) ? KAC : 0;
                int q1 = base + j + 1; q1 -= (q1 >= KAC) ? KAC : 0;
                int q2 = base + j + 2; q2 -= (q2 >= KAC) ? KAC : 0;
                int q3 = base + j + 3; q3 -= (q3 >= KAC) ? KAC : 0;
                s0 = fmaf(ackS[tid][j    ], gring[q0][tid], s0);
                s1 = fmaf(ackS[tid][j + 1], gring[q1][tid], s1);
                s2 = fmaf(ackS[tid][j + 2], gring[q2][tid], s2);
                s3 = fmaf(ackS[tid][j + 3], gring[q3][tid], s3);
            }
            const float s = (s0 + s1) + (s2 + s3);
            aclS[tid] = 1.0f / (1.0f + __expf(-slA * (s - ofA)));
        }
        __syncthreads();
        // P2a (tid<14): feedback via Wcomb, release prob, vesicle pools
        if (tid < NUM_BC) {
            float f0 = 0.f, f1 = 0.f, f2 = 0.f, f3 = 0.f;
            #pragma unroll
            for (int a = 0; a < 44; a += 4) {
                f0 = fmaf(wcbS[tid][a    ], aclS[a    ], f0);
                f1 = fmaf(wcbS[tid][a + 1], aclS[a + 1], f1);
                f2 = fmaf(wcbS[tid][a + 2], aclS[a + 2], f2);
                f3 = fmaf(wcbS[tid][a + 3], aclS[a + 3], f3);
            }
            f0 = fmaf(wcbS[tid][44], aclS[44], f0);
            const float fb = (f0 + f1) + (f2 + f3);
            const float d  = drive[(size_t)i * DRV_STRIDE + tid];
            const float rp = 1.0f / (1.0f + __expf(-slB * (d + fb - ofB)));
            const float release = rrp * rp;
            const float mv12 = p12 * ip * (1.0f - rrp / rrc);
            const float mv01 = p01 * (ipc - ip);
            rrp = rrp - release + mv12;
            ip  = ip - mv12 + mv01;
            relS[tid] = release;
            relout[(size_t)i * NUM_BC + tid] = release;
        }
        // P2b (tid 14..58, concurrent): aclO = aclS + W_aa @ aclS -> history
        if (tid >= NUM_BC && aout < NUM_ACL) {
            float o0 = aclS[aout], o1 = 0.f, o2 = 0.f, o3 = 0.f;
            #pragma unroll
            for (int a2 = 0; a2 < 44; a2 += 4) {
                o0 = fmaf(waaS[aout][a2    ], aclS[a2    ], o0);
                o1 = fmaf(waaS[aout][a2 + 1], aclS[a2 + 1], o1);
                o2 = fmaf(waaS[aout][a2 + 2], aclS[a2 + 2], o2);
                o3 = fmaf(waaS[aout][a2 + 3], aclS[a2 + 3], o3);
            }
            o0 = fmaf(waaS[aout][44], aclS[44], o0);
            aclout[(size_t)i * NUM_ACL + aout] = (o0 + o1) + (o2 + o3);
        }
        __syncthreads();
        // P3: g[t=i] = W_ba @ release; overwrite oldest ring slot
        if (tid < NUM_ACL) {
            float g0 = 0.f, g1 = 0.f;
            #pragma unroll
            for (int b = 0; b < NUM_BC; b += 2) {
                g0 = fmaf(wbaS[tid][b    ], relS[b    ], g0);
                g1 = fmaf(wbaS[tid][b + 1], relS[b + 1], g1);
            }
            gring[base][tid] = g0 + g1;
        }
        __syncthreads();
    }
}

// =================================================================
// Kernel 3: sensor FIRs. Each block stages one channel's column
// (319 samples) into LDS via gfx1250 async global->LDS, then runs
// the 64-tap correlation entirely out of LDS.
// =================================================================
#define SENS_T 256
__global__ __launch_bounds__(256) void sensor_kernel(
    const float* __restrict__ ws, float* __restrict__ out)
{
    __shared__ float col[SENS_T + K1];       // 320 floats
    __shared__ float ksh[K1];
    const int tid = threadIdx.x;
    const int t0  = blockIdx.x * SENS_T;
    const int ch  = blockIdx.y;
    const int nload = SENS_T + K1 - 1;       // 319; max t0+j = 32830 = TP-1

    const float* __restrict__ src;
    const float* __restrict__ kern;
    int stride, chan; size_t obase;
    if (ch < NUM_BC) {
        src = out + OUT_REL_OFF; kern = ws + OFF_IGLU;
        stride = NUM_BC; chan = ch; obase = 0;
    } else {
        src = ws + OFF_ACL; kern = ws + OFF_GCAMP;
        stride = NUM_ACL; chan = ch - NUM_BC; obase = OUT_YC_OFF;
    }

    if (tid < K1) ksh[tid] = kern[tid];
#ifdef HAVE_ASYNC_LDS
    for (int j = tid; j < nload; j += SENS_T) {
        ASYNC_CP_F32(src + (size_t)(t0 + j) * stride + chan, &col[j]);
    }
    WAIT_ASYNC();
    __syncthreads();
#else
    for (int j = tid; j < nload; j += SENS_T)
        col[j] = src[(size_t)(t0 + j) * stride + chan];
    __syncthreads();
#endif

    const int t = t0 + tid;
    if (t >= T_STIM) return;
    float a0 = 0.f, a1 = 0.f, a2 = 0.f, a3 = 0.f;
    #pragma unroll
    for (int k = 0; k < K1; k += 4) {
        a0 = fmaf(col[tid + k    ], ksh[k    ], a0);
        a1 = fmaf(col[tid + k + 1], ksh[k + 1], a1);
        a2 = fmaf(col[tid + k + 2], ksh[k + 2], a2);
        a3 = fmaf(col[tid + k + 3], ksh[k + 3], a3);
    }
    out[obase + (size_t)t * stride + chan] = (a0 + a1) + (a2 + a3);
}

// =================================================================
extern "C" void kernel_launch(void* const* d_in, const int* in_sizes, int n_in,
                              void* d_out, int out_size, void* d_ws, size_t ws_size,
                              hipStream_t stream)
{
    const float* x     = (const float*)d_in[0];
    const float* lw    = (const float*)d_in[1];
    const float* sb    = (const float*)d_in[2];
    const float* ssc   = (const float*)d_in[3];
    const float* lks   = (const float*)d_in[4];
    const float* soff  = (const float*)d_in[5];
    const float* lss   = (const float*)d_in[6];
    const float* lcp01 = (const float*)d_in[7];
    const float* lcp12 = (const float*)d_in[8];
    const float* lic   = (const float*)d_in[9];
    const float* lrc   = (const float*)d_in[10];
    const float* sips  = (const float*)d_in[11];
    const float* srps  = (const float*)d_in[12];
    const float* lba   = (const float*)d_in[13];
    const float* ltr   = (const float*)d_in[14];
    const float* ltd   = (const float*)d_in[15];
    const float* lass  = (const float*)d_in[16];
    const float* aso   = (const float*)d_in[17];
    const float* lab   = (const float*)d_in[18];
    const float* laa   = (const float*)d_in[19];
    float* ws  = (float*)d_ws;
    float* out = (float*)d_out;

    hipLaunchKernelGGL(precompute_kernel, dim3(1), dim3(128), 0, stream,
        lw, lks, soff, lss, lcp01, lcp12, lic, lrc, sips, srps,
        lba, ltr, ltd, lass, aso, lab, laa, ws);

    const int ntiles = (TP + 15) / 16;
    hipLaunchKernelGGL(drive_wmma_kernel, dim3((ntiles + 7) / 8), dim3(256), 0, stream,
        x, sb, ssc, ws);

    hipLaunchKernelGGL(scan_kernel, dim3(1), dim3(64), 0, stream, ws, out);

    hipLaunchKernelGGL(sensor_kernel, dim3((T_STIM + SENS_T - 1) / SENS_T, NUM_BC + NUM_ACL),
        dim3(256), 0, stream, ws, out);
}